// XYPolynomial_70901320122523
// MI455X (gfx1250) — compile-verified
//
#include <hip/hip_runtime.h>
#include <hip/hip_bf16.h>
#include <stdint.h>

#define TPB   256                 // threads per block (8 wave32)
#define TILE2 (TPB * 2)           // 512 rays per block, 2 rays/lane packed fp32
#define N_ITER 10
#define DAMPING 0.95f
#define TOL 1e-6f
#define APERTURE_SLACK 1e-3f

typedef float v2f __attribute__((ext_vector_type(2)));

__device__ __forceinline__ v2f v2splat(float s) { v2f r; r.x = s; r.y = s; return r; }
__device__ __forceinline__ v2f v2div(v2f a, v2f b) { v2f r; r.x = a.x / b.x; r.y = a.y / b.y; return r; }

// ---- forced CDNA5 packed-FP32 VOP3P ops (2 rays per instruction) ----
__device__ __forceinline__ v2f pk_fma(v2f a, v2f b, v2f c) {
    v2f d; asm("v_pk_fma_f32 %0, %1, %2, %3" : "=v"(d) : "v"(a), "v"(b), "v"(c)); return d;
}
__device__ __forceinline__ v2f pk_mul(v2f a, v2f b) {
    v2f d; asm("v_pk_mul_f32 %0, %1, %2" : "=v"(d) : "v"(a), "v"(b)); return d;
}
__device__ __forceinline__ v2f pk_add(v2f a, v2f b) {
    v2f d; asm("v_pk_add_f32 %0, %1, %2" : "=v"(d) : "v"(a), "v"(b)); return d;
}

// rcp (v_rcp_f32 TRANS) + one packed Newton-Raphson refinement (~1 ulp)
__device__ __forceinline__ v2f v2rcp_fast(v2f b, v2f one, v2f mone) {
    v2f r; r.x = __builtin_amdgcn_rcpf(b.x); r.y = __builtin_amdgcn_rcpf(b.y);
    v2f nb = pk_mul(b, mone);
    v2f e  = pk_fma(nb, r, one);     // e = 1 - b*r
    return pk_fma(r, e, r);          // r' = r + r*e
}

// -------- packed sag + gradient: conic + 6x6 poly via double-Horner --------
// cc[36] = splatted coefficients; mk=-(1+K)C^2, ck=C*(1+K)*C^2
__device__ __forceinline__ void sag_eval2(v2f x, v2f y, const v2f* __restrict__ cc,
                                          v2f one, v2f mone, v2f two, v2f half,
                                          v2f cC, v2f mk, v2f ck, v2f scl2,
                                          v2f& g, v2f& gx, v2f& gy, v2f& u2v)
{
    // conical part
    v2f r2 = pk_fma(x, x, pk_mul(y, y));
    v2f u2 = pk_fma(r2, mk, one);
    u2v = u2;
    v2f u2c; u2c.x = fmaxf(u2.x, 1e-12f); u2c.y = fmaxf(u2.y, 1e-12f);
    v2f u;  u.x  = __builtin_amdgcn_sqrtf(u2c.x); u.y  = __builtin_amdgcn_sqrtf(u2c.y);
    v2f ru; ru.x = __builtin_amdgcn_rsqf(u2c.x);  ru.y = __builtin_amdgcn_rsqf(u2c.y);   // 1/u
    v2f op  = pk_add(u, one);
    v2f iop = v2rcp_fast(op, one, mone);           // 1/(1+u)
    v2f g1  = pk_mul(pk_mul(r2, iop), cC);
    // dg = C*iop + (C*r2*(1+K)*C^2) * iop^2 * (0.5/u)
    v2f num  = pk_mul(r2, ck);
    v2f corr = pk_mul(pk_mul(iop, iop), pk_mul(num, pk_mul(ru, half)));
    v2f dg   = pk_fma(cC, iop, corr);
    v2f gx1 = pk_mul(pk_mul(x, dg), two);
    v2f gy1 = pk_mul(pk_mul(y, dg), two);

    // rows: P_p(y) and P_p'(y) by double-Horner (5 + 4 fma per row)
    v2f cy[6], cyd[6];
#pragma unroll
    for (int p = 0; p < 6; ++p) {
        const v2f* c = cc + p * 6;
        v2f b = c[5];
        v2f d = c[5];
        b = pk_fma(b, y, c[4]);  d = pk_fma(d, y, b);
        b = pk_fma(b, y, c[3]);  d = pk_fma(d, y, b);
        b = pk_fma(b, y, c[2]);  d = pk_fma(d, y, b);
        b = pk_fma(b, y, c[1]);  d = pk_fma(d, y, b);
        b = pk_fma(b, y, c[0]);
        cy[p] = b; cyd[p] = d;
    }
    // columns: g2 = H_x(cy), gx2 = H_x'(cy), gy2 = H_x(cyd)
    v2f e = cy[5], f = cy[5];
    e = pk_fma(e, x, cy[4]);  f = pk_fma(f, x, e);
    e = pk_fma(e, x, cy[3]);  f = pk_fma(f, x, e);
    e = pk_fma(e, x, cy[2]);  f = pk_fma(f, x, e);
    e = pk_fma(e, x, cy[1]);  f = pk_fma(f, x, e);
    e = pk_fma(e, x, cy[0]);
    v2f h = cyd[5];
    h = pk_fma(h, x, cyd[4]);
    h = pk_fma(h, x, cyd[3]);
    h = pk_fma(h, x, cyd[2]);
    h = pk_fma(h, x, cyd[1]);
    h = pk_fma(h, x, cyd[0]);

    g  = pk_mul(pk_add(g1, e), scl2);
    gx = pk_mul(pk_add(gx1, f), scl2);
    gy = pk_mul(pk_add(gy1, h), scl2);
}

// -------- main trace kernel (defined first so disasm snippet shows it) --------
__global__ __launch_bounds__(TPB) void trace_kernel(
    const float* __restrict__ P, const float* __restrict__ V,
    const float* __restrict__ tf, const float* __restrict__ dia,
    const float* __restrict__ Cp, const float* __restrict__ Kp,
    const float* __restrict__ coeff, const float* __restrict__ scalep,
    const float* __restrict__ Minv, float* __restrict__ out, int N)
{
    __shared__ float sh[6 * TILE2];                   // [0..1535]=P tile, [1536..3071]=V tile
    const int       tid    = threadIdx.x;
    const long long base   = (long long)blockIdx.x * TILE2;
    const int       remain = (int)((long long)N - base);
    const long long i0     = base + tid;
    const long long i1     = base + TPB + tid;

    v2f px = v2splat(0.f), py = v2splat(0.f), pz = v2splat(-10.f);
    v2f vx = v2splat(0.f), vy = v2splat(0.f), vz = v2splat(1.f);

    if (remain >= TILE2) {
        // Async global -> LDS staging: 2 x 6144 B per tile, 16B granules (ASYNCcnt path).
        const char* gp = (const char*)(P + base * 3);
        const char* gv = (const char*)(V + base * 3);
#pragma unroll 1
        for (int c = tid; c < 384; c += TPB) {        // wave-uniform trip counts
            const unsigned off = (unsigned)c * 16u;
            const unsigned lds = (unsigned)(size_t)(&sh[0]) + off;
            asm volatile("global_load_async_to_lds_b128 %0, %1, %2"
                         :: "v"(lds), "v"(off), "s"(gp) : "memory");
        }
#pragma unroll 1
        for (int c = tid; c < 384; c += TPB) {
            const unsigned off = (unsigned)c * 16u;
            const unsigned lds = (unsigned)(size_t)(&sh[1536]) + off;
            asm volatile("global_load_async_to_lds_b128 %0, %1, %2"
                         :: "v"(lds), "v"(off), "s"(gv) : "memory");
        }
        asm volatile("s_wait_asynccnt 0x0" ::: "memory");
        __syncthreads();
        const int r0 = tid, r1 = tid + TPB;
        px.x = sh[r0 * 3 + 0]; py.x = sh[r0 * 3 + 1]; pz.x = sh[r0 * 3 + 2];
        px.y = sh[r1 * 3 + 0]; py.y = sh[r1 * 3 + 1]; pz.y = sh[r1 * 3 + 2];
        vx.x = sh[1536 + r0 * 3 + 0]; vy.x = sh[1536 + r0 * 3 + 1]; vz.x = sh[1536 + r0 * 3 + 2];
        vx.y = sh[1536 + r1 * 3 + 0]; vy.y = sh[1536 + r1 * 3 + 1]; vz.y = sh[1536 + r1 * 3 + 2];
    } else {                                          // ragged tail block: direct loads
        if (tid < remain) {
            px.x = P[i0 * 3 + 0]; py.x = P[i0 * 3 + 1]; pz.x = P[i0 * 3 + 2];
            vx.x = V[i0 * 3 + 0]; vy.x = V[i0 * 3 + 1]; vz.x = V[i0 * 3 + 2];
        }
        if (tid + TPB < remain) {
            px.y = P[i1 * 3 + 0]; py.y = P[i1 * 3 + 1]; pz.y = P[i1 * 3 + 2];
            vx.y = V[i1 * 3 + 0]; vy.y = V[i1 * 3 + 1]; vz.y = V[i1 * 3 + 2];
        }
    }

    // Uniform constants -> SMEM scalar loads
    const float C = Cp[0], K = Kp[0], scl = scalep[0];
    const float rmax2 = 0.25f * dia[0] * dia[0];
    const float c2k1  = (1.0f + K) * C * C;
    const float m00 = Minv[0], m01 = Minv[1], m02 = Minv[2],  m03 = Minv[3];
    const float m10 = Minv[4], m11 = Minv[5], m12 = Minv[6],  m13 = Minv[7];
    const float m20 = Minv[8], m21 = Minv[9], m22 = Minv[10], m23 = Minv[11];

    // Splatted loop-invariant operands for packed ops
    v2f cc[36];
#pragma unroll
    for (int j = 0; j < 36; ++j) cc[j] = v2splat(coeff[j]);
    const v2f one  = v2splat(1.0f), mone = v2splat(-1.0f);
    const v2f two  = v2splat(2.0f), half = v2splat(0.5f);
    const v2f cC   = v2splat(C),    mk   = v2splat(-c2k1);
    const v2f ck   = v2splat(C * c2k1), scl2 = v2splat(scl);

    // Local frame (packed)
    v2f Plx = pk_fma(v2splat(m00), px, pk_fma(v2splat(m01), py, pk_fma(v2splat(m02), pz, v2splat(m03))));
    v2f Ply = pk_fma(v2splat(m10), px, pk_fma(v2splat(m11), py, pk_fma(v2splat(m12), pz, v2splat(m13))));
    v2f Plz = pk_fma(v2splat(m20), px, pk_fma(v2splat(m21), py, pk_fma(v2splat(m22), pz, v2splat(m23))));
    v2f Vlx = pk_fma(v2splat(m00), vx, pk_fma(v2splat(m01), vy, pk_mul(vz, v2splat(m02))));
    v2f Vly = pk_fma(v2splat(m10), vx, pk_fma(v2splat(m11), vy, pk_mul(vz, v2splat(m12))));
    v2f Vlz = pk_fma(v2splat(m20), vx, pk_fma(v2splat(m21), vy, pk_mul(vz, v2splat(m22))));
    v2f negVlz = pk_mul(Vlz, mone);

    v2f t = v2div(-Plz, Vlz);                         // once: keep precise

    // Damped Newton, 10 iterations, packed fp32
#pragma unroll 1
    for (int it = 0; it < N_ITER; ++it) {
        v2f Qx = pk_fma(t, Vlx, Plx);
        v2f Qy = pk_fma(t, Vly, Ply);
        v2f Qz = pk_fma(t, Vlz, Plz);
        v2f g, gx, gy, u2v;
        sag_eval2(Qx, Qy, cc, one, mone, two, half, cC, mk, ck, scl2, g, gx, gy, u2v);
        v2f F  = g - Qz;
        v2f dF = pk_fma(gx, Vlx, pk_fma(gy, Vly, negVlz));
        dF.x = (fabsf(dF.x) < 1e-9f) ? ((dF.x >= 0.0f) ? 1e-9f : -1e-9f) : dF.x;
        dF.y = (fabsf(dF.y) < 1e-9f) ? ((dF.y >= 0.0f) ? 1e-9f : -1e-9f) : dF.y;
        v2f q = pk_mul(F, v2rcp_fast(dF, one, mone)); // fast F/dF (Newton self-corrects)
        t.x = (fabsf(F.x) < TOL) ? t.x : fmaf(-DAMPING, q.x, t.x);
        t.y = (fabsf(F.y) < TOL) ? t.y : fmaf(-DAMPING, q.y, t.y);
    }

    // Final evaluation + normals + aperture test
    v2f Qx = pk_fma(t, Vlx, Plx);
    v2f Qy = pk_fma(t, Vly, Ply);
    v2f Qz = pk_fma(t, Vlz, Plz);
    v2f g, gx, gy, u2v;
    sag_eval2(Qx, Qy, cc, one, mone, two, half, cC, mk, ck, scl2, g, gx, gy, u2v);
    v2f F = g - Qz;

    v2f nn = pk_fma(gx, gx, pk_fma(gy, gy, one));
    v2f inr; inr.x = 1.0f / sqrtf(nn.x); inr.y = 1.0f / sqrtf(nn.y);   // once: precise
    v2f nlx = pk_mul(gx, inr), nly = pk_mul(gy, inr);
    v2f nlz = pk_mul(inr, mone);
    const float t00 = tf[0], t01 = tf[1], t02 = tf[2];
    const float t10 = tf[4], t11 = tf[5], t12 = tf[6];
    const float t20 = tf[8], t21 = tf[9], t22 = tf[10];
    v2f nx = pk_fma(v2splat(t00), nlx, pk_fma(v2splat(t01), nly, pk_mul(nlz, v2splat(t02))));
    v2f ny = pk_fma(v2splat(t10), nlx, pk_fma(v2splat(t11), nly, pk_mul(nlz, v2splat(t12))));
    v2f nz = pk_fma(v2splat(t20), nlx, pk_fma(v2splat(t21), nly, pk_mul(nlz, v2splat(t22))));

    v2f r2q = pk_fma(Qx, Qx, pk_mul(Qy, Qy));
    float valid0 = ((u2v.x > 0.0f) && (r2q.x <= rmax2) && (fabsf(F.x) < APERTURE_SLACK)) ? 1.0f : 0.0f;
    float valid1 = ((u2v.y > 0.0f) && (r2q.y <= rmax2) && (fabsf(F.y) < APERTURE_SLACK)) ? 1.0f : 0.0f;

    const long long NN = (long long)N;
    if (tid < remain) {
        out[i0]              = t.x;
        out[NN + 3 * i0 + 0] = nx.x;
        out[NN + 3 * i0 + 1] = ny.x;
        out[NN + 3 * i0 + 2] = nz.x;
        out[4 * NN + i0]     = valid0;
    }
    if (tid + TPB < remain) {
        out[i1]              = t.y;
        out[NN + 3 * i1 + 0] = nx.y;
        out[NN + 3 * i1 + 1] = ny.y;
        out[NN + 3 * i1 + 2] = nz.y;
        out[4 * NN + i1]     = valid1;
    }
}

// -------- setup: invert tf (adjugate, double), write tf tails --------
__global__ void setup_kernel(const float* __restrict__ tf,
                             float* __restrict__ Minv,
                             float* __restrict__ tail)
{
    if (threadIdx.x != 0 || blockIdx.x != 0) return;
    double m[16], inv[16];
    for (int i = 0; i < 16; ++i) m[i] = (double)tf[i];
    inv[0]  =  m[5]*m[10]*m[15] - m[5]*m[11]*m[14] - m[9]*m[6]*m[15] + m[9]*m[7]*m[14] + m[13]*m[6]*m[11] - m[13]*m[7]*m[10];
    inv[4]  = -m[4]*m[10]*m[15] + m[4]*m[11]*m[14] + m[8]*m[6]*m[15] - m[8]*m[7]*m[14] - m[12]*m[6]*m[11] + m[12]*m[7]*m[10];
    inv[8]  =  m[4]*m[9]*m[15]  - m[4]*m[11]*m[13] - m[8]*m[5]*m[15] + m[8]*m[7]*m[13] + m[12]*m[5]*m[11] - m[12]*m[7]*m[9];
    inv[12] = -m[4]*m[9]*m[14]  + m[4]*m[10]*m[13] + m[8]*m[5]*m[14] - m[8]*m[6]*m[13] - m[12]*m[5]*m[10] + m[12]*m[6]*m[9];
    inv[1]  = -m[1]*m[10]*m[15] + m[1]*m[11]*m[14] + m[9]*m[2]*m[15] - m[9]*m[3]*m[14] - m[13]*m[2]*m[11] + m[13]*m[3]*m[10];
    inv[5]  =  m[0]*m[10]*m[15] - m[0]*m[11]*m[14] - m[8]*m[2]*m[15] + m[8]*m[3]*m[14] + m[12]*m[2]*m[11] - m[12]*m[3]*m[10];
    inv[9]  = -m[0]*m[9]*m[15]  + m[0]*m[11]*m[13] + m[8]*m[1]*m[15] - m[8]*m[3]*m[13] - m[12]*m[1]*m[11] + m[12]*m[3]*m[9];
    inv[13] =  m[0]*m[9]*m[14]  - m[0]*m[10]*m[13] - m[8]*m[1]*m[14] + m[8]*m[2]*m[13] + m[12]*m[1]*m[10] - m[12]*m[2]*m[9];
    inv[2]  =  m[1]*m[6]*m[15]  - m[1]*m[7]*m[14]  - m[5]*m[2]*m[15] + m[5]*m[3]*m[14] + m[13]*m[2]*m[7]  - m[13]*m[3]*m[6];
    inv[6]  = -m[0]*m[6]*m[15]  + m[0]*m[7]*m[14]  + m[4]*m[2]*m[15] - m[4]*m[3]*m[14] - m[12]*m[2]*m[7]  + m[12]*m[3]*m[6];
    inv[10] =  m[0]*m[5]*m[15]  - m[0]*m[7]*m[13]  - m[4]*m[1]*m[15] + m[4]*m[3]*m[13] + m[12]*m[1]*m[7]  - m[12]*m[3]*m[5];
    inv[14] = -m[0]*m[5]*m[14]  + m[0]*m[6]*m[13]  + m[4]*m[1]*m[14] - m[4]*m[2]*m[13] - m[12]*m[1]*m[6]  + m[12]*m[2]*m[5];
    inv[3]  = -m[1]*m[6]*m[11]  + m[1]*m[7]*m[10]  + m[5]*m[2]*m[11] - m[5]*m[3]*m[10] - m[9]*m[2]*m[7]   + m[9]*m[3]*m[6];
    inv[7]  =  m[0]*m[6]*m[11]  - m[0]*m[7]*m[10]  - m[4]*m[2]*m[11] + m[4]*m[3]*m[10] + m[8]*m[2]*m[7]   - m[8]*m[3]*m[6];
    inv[11] = -m[0]*m[5]*m[11]  + m[0]*m[7]*m[9]   + m[4]*m[1]*m[11] - m[4]*m[3]*m[9]  - m[8]*m[1]*m[7]   + m[8]*m[3]*m[5];
    inv[15] =  m[0]*m[5]*m[10]  - m[0]*m[6]*m[9]   - m[4]*m[1]*m[10] + m[4]*m[2]*m[9]  + m[8]*m[1]*m[6]   - m[8]*m[2]*m[5];
    double det = m[0]*inv[0] + m[1]*inv[4] + m[2]*inv[8] + m[3]*inv[12];
    double id  = 1.0 / det;
    for (int i = 0; i < 16; ++i) Minv[i] = (float)(inv[i] * id);
    for (int i = 0; i < 16; ++i) { tail[i] = tf[i]; tail[16 + i] = tf[i]; }
}

extern "C" void kernel_launch(void* const* d_in, const int* in_sizes, int n_in,
                              void* d_out, int out_size, void* d_ws, size_t ws_size,
                              hipStream_t stream)
{
    const float* P     = (const float*)d_in[0];
    const float* V     = (const float*)d_in[1];
    const float* tf    = (const float*)d_in[2];
    const float* dia   = (const float*)d_in[3];
    const float* C     = (const float*)d_in[4];
    const float* K     = (const float*)d_in[5];
    const float* coeff = (const float*)d_in[6];
    const float* scale = (const float*)d_in[7];
    const int    N     = in_sizes[0] / 3;

    float* out  = (float*)d_out;
    float* Minv = (float*)d_ws;          // 16 floats of scratch

    hipLaunchKernelGGL(setup_kernel, dim3(1), dim3(32), 0, stream,
                       tf, Minv, out + 5LL * (long long)N);

    const int blocks = (N + TILE2 - 1) / TILE2;
    hipLaunchKernelGGL(trace_kernel, dim3(blocks), dim3(TPB), 0, stream,
                       P, V, tf, dia, C, K, coeff, scale, Minv, out, N);
}